// DynamicCenterLoss_12807592477411
// MI455X (gfx1250) — compile-verified
//
#include <hip/hip_runtime.h>
#include <hip/hip_bf16.h>

namespace {
constexpr int D = 64;        // feature dim
constexpr int C = 13;        // classes
constexpr int BATCHES = 8;   // point clouds
constexpr int IGN = -1;
constexpr float MARGIN = 0.5f;
constexpr float LOSS_W = 0.01f;
}

typedef __attribute__((ext_vector_type(2))) float v2f;
typedef __attribute__((ext_vector_type(8))) float v8f;

__global__ void dcl_zero_ws(float* ws, int n) {
    int i = blockIdx.x * blockDim.x + threadIdx.x;
    if (i < n) ws[i] = 0.f;
}

// Flush per-wave WMMA accumulators (class sums for batch cur_b), per-lane class
// counts, and per-lane intra-d2 partials into the block LDS accumulators.
__device__ __forceinline__ void wave_flush(int cur_b, v8f& a0, v8f& a1, v8f& a2, v8f& a3,
                                           float& cntacc, float& d2acc,
                                           float* s_fsum, float* s_ccnt, float* s_d2,
                                           int lane) {
    if (cur_b < 0) return;
    const int half = lane >> 4;
    const int m = lane & 15;
    // intra d2: full-wave reduce of per-lane partials
    float pd = d2acc;
#pragma unroll
    for (int off = 16; off >= 1; off >>= 1) pd += __shfl_xor(pd, off, 32);
    if (lane == 0 && pd != 0.f) atomicAdd(&s_d2[cur_b], pd);
    // class counts: lane m of each half counted disjoint points of class m
    if (m < C && cntacc != 0.f) atomicAdd(&s_ccnt[cur_b * C + m], cntacc);
    // C/D fragment layout: lane l, vgpr v -> (class = v + 8*(l>=16), feature = 16*nb + (l&15))
    float* bbase = s_fsum + (size_t)cur_b * C * D;
#pragma unroll
    for (int v = 0; v < 8; ++v) {
        int cls = v + (half << 3);
        if (cls < C) {
            float* rp = bbase + cls * D + m;
            atomicAdd(rp + 0,  a0[v]);
            atomicAdd(rp + 16, a1[v]);
            atomicAdd(rp + 32, a2[v]);
            atomicAdd(rp + 48, a3[v]);
        }
    }
    const v8f vz = {};
    a0 = vz; a1 = vz; a2 = vz; a3 = vz;
    cntacc = 0.f;
    d2acc = 0.f;
}

// Scalar fallback: one point, whole wave cooperates (lane = feature, feature+32).
__device__ __forceinline__ void scalar_point(int p, const int* target, const int* batch,
                                             const float* feat, const float* s_cent,
                                             float* s_fsum, float* s_ccnt, float* s_d2,
                                             int lane) {
    int t = target[p];              // uniform load
    if (t == IGN) return;           // uniform branch
    int b = batch[p];
    const float* row = feat + (size_t)p * D;
    float x0 = row[lane], x1 = row[lane + 32];
    float* fb = s_fsum + ((size_t)b * C + t) * D;
    atomicAdd(fb + lane, x0);
    atomicAdd(fb + lane + 32, x1);
    float c0 = s_cent[t * D + lane], c1 = s_cent[t * D + lane + 32];
    float d0 = x0 - c0, d1 = x1 - c1;
    float pd = d0 * d0 + d1 * d1;
#pragma unroll
    for (int off = 16; off >= 1; off >>= 1) pd += __shfl_xor(pd, off, 32);
    if (lane == 0) {
        atomicAdd(&s_d2[b], pd);
        atomicAdd(&s_ccnt[b * C + t], 1.f);
    }
}

__global__ __launch_bounds__(256) void dcl_phase1(
    const int* __restrict__ target, const float* __restrict__ feat,
    const int* __restrict__ batch, const float* __restrict__ centers,
    float* __restrict__ g_fsum, float* __restrict__ g_ccnt, float* __restrict__ g_d2,
    int nPoints, int pointsPerWave) {
    __shared__ float s_cent[C * D];            // 3.3 KB cached EMA centers
    __shared__ float s_fsum[BATCHES * C * D];  // 26 KB per-block segment sums
    __shared__ float s_ccnt[BATCHES * C];
    __shared__ float s_d2[BATCHES];

    for (int i = threadIdx.x; i < C * D; i += blockDim.x) s_cent[i] = centers[i];
    for (int i = threadIdx.x; i < BATCHES * C * D; i += blockDim.x) s_fsum[i] = 0.f;
    for (int i = threadIdx.x; i < BATCHES * C; i += blockDim.x) s_ccnt[i] = 0.f;
    if (threadIdx.x < BATCHES) s_d2[threadIdx.x] = 0.f;
    __syncthreads();

    const int lane = threadIdx.x & 31;
    const int half = lane >> 4;     // 0: points {0,1} of group, 1: points {2,3}
    const int m = lane & 15;        // class row / feature column within 16-block
    const int wavesPerBlock = blockDim.x >> 5;
    const int waveId = blockIdx.x * wavesPerBlock + (threadIdx.x >> 5);

    const int start = waveId * pointsPerWave;
    const int stop = min(start + pointsPerWave, nPoints);

    v8f acc0 = {}, acc1 = {}, acc2 = {}, acc3 = {};  // 16x16 f32 C/D frags, feature blocks 0..3
    float cntacc = 0.f;   // per-lane class-count partial (from one-hot A frags)
    float d2acc = 0.f;    // per-lane intra-d2 partial
    int cur_b = -1;

    // Supergroups of 16 points: batch the global loads for deep MLP.
    for (int sg = start; sg < stop; sg += 16) {
        if (sg + 16 <= nPoints) {
            int4 tt[4], bt[4];
#pragma unroll
            for (int g = 0; g < 4; ++g) {
                tt[g] = *(const int4*)(target + sg + 4 * g);
                bt[g] = *(const int4*)(batch + sg + 4 * g);
            }
            // B fragments (4x16, K=points, N=features): lanes 0-15 hold K={0,1},
            // lanes 16-31 hold K={2,3}; column = 16*nb + (lane&15).
            v2f bf[4][4];
#pragma unroll
            for (int g = 0; g < 4; ++g) {
                const float* rA = feat + (size_t)(sg + 4 * g + 2 * half) * D;
#pragma unroll
                for (int nb = 0; nb < 4; ++nb) {
                    int col = nb * 16 + m;
                    bf[g][nb].x = rA[col];
                    bf[g][nb].y = rA[D + col];
                }
            }
#pragma unroll
            for (int g = 0; g < 4; ++g) {
                int t0 = tt[g].x, t1 = tt[g].y, t2 = tt[g].z, t3 = tt[g].w;
                int b0 = bt[g].x, b1 = bt[g].y, b2 = bt[g].z, b3 = bt[g].w;
                bool uni = (b0 == b1) && (b1 == b2) && (b2 == b3);  // lane-uniform
                if (uni) {
                    if (b0 != cur_b) {
                        wave_flush(cur_b, acc0, acc1, acc2, acc3, cntacc, d2acc,
                                   s_fsum, s_ccnt, s_d2, lane);
                        cur_b = b0;
                    }
                    int ta = half ? t2 : t0;
                    int tb = half ? t3 : t1;
                    bool va = (ta != IGN), vb = (tb != IGN);
                    int ca = va ? ta : 0, cb = vb ? tb : 0;
                    // One-hot A fragment (16x4, M=class, K=points)
                    v2f a;
                    a.x = (va && ta == m) ? 1.f : 0.f;
                    a.y = (vb && tb == m) ? 1.f : 0.f;
                    cntacc += a.x + a.y;
                    // intra d2 from already-loaded fragments vs cached centers
#pragma unroll
                    for (int nb = 0; nb < 4; ++nb) {
                        int col = nb * 16 + m;
                        float da = bf[g][nb].x - s_cent[ca * D + col];
                        float db = bf[g][nb].y - s_cent[cb * D + col];
                        d2acc += (va ? da * da : 0.f) + (vb ? db * db : 0.f);
                    }
                    // one-hot GEMM: per-class feature sums, full f32
                    acc0 = __builtin_amdgcn_wmma_f32_16x16x4_f32(false, a, false, bf[g][0], (short)0, acc0, false, false);
                    acc1 = __builtin_amdgcn_wmma_f32_16x16x4_f32(false, a, false, bf[g][1], (short)0, acc1, false, false);
                    acc2 = __builtin_amdgcn_wmma_f32_16x16x4_f32(false, a, false, bf[g][2], (short)0, acc2, false, false);
                    acc3 = __builtin_amdgcn_wmma_f32_16x16x4_f32(false, a, false, bf[g][3], (short)0, acc3, false, false);
                } else {
                    // batch boundary inside group (rare: batch is sorted)
#pragma unroll
                    for (int k = 0; k < 4; ++k)
                        scalar_point(sg + 4 * g + k, target, batch, feat, s_cent,
                                     s_fsum, s_ccnt, s_d2, lane);
                }
            }
        } else {
            for (int p = sg; p < stop; ++p)
                scalar_point(p, target, batch, feat, s_cent, s_fsum, s_ccnt, s_d2, lane);
        }
    }
    wave_flush(cur_b, acc0, acc1, acc2, acc3, cntacc, d2acc, s_fsum, s_ccnt, s_d2, lane);

    __syncthreads();

    // Block -> global flush, restricted to the batch range this contiguous chunk
    // touched (batch is sorted, so usually 1-2 of 8 batches -> ~4-8x fewer atomics).
    const int blkStart = blockIdx.x * wavesPerBlock * pointsPerWave;
    const int blkStop = min(blkStart + wavesPerBlock * pointsPerWave, nPoints);
    if (blkStart < nPoints) {
        const int bmin = batch[blkStart];
        const int bmax = batch[blkStop - 1];
        const int fbase = bmin * C * D;
        const int fcnt = (bmax - bmin + 1) * C * D;
        for (int i = threadIdx.x; i < fcnt; i += blockDim.x) {
            float v = s_fsum[fbase + i];
            if (v != 0.f) atomicAdd(&g_fsum[fbase + i], v);
        }
        const int cbase = bmin * C;
        const int ccnt = (bmax - bmin + 1) * C;
        for (int i = threadIdx.x; i < ccnt; i += blockDim.x) {
            float v = s_ccnt[cbase + i];
            if (v != 0.f) atomicAdd(&g_ccnt[cbase + i], v);
        }
        if ((int)threadIdx.x <= bmax - bmin) {
            int b = bmin + threadIdx.x;
            if (s_d2[b] != 0.f) atomicAdd(&g_d2[b], s_d2[b]);
        }
    }
}

__global__ __launch_bounds__(256) void dcl_finalize(
    const float* __restrict__ g_fsum, const float* __restrict__ g_ccnt,
    const float* __restrict__ g_d2, float* __restrict__ out) {
    __shared__ float s_cm[BATCHES][16][D + 1];   // class means, padded rows (bank-friendly)
    __shared__ float s_gram[BATCHES][16][16];    // cm . cm^T per batch
    __shared__ float s_inter[BATCHES];
    __shared__ float s_np[BATCHES];

    for (int i = threadIdx.x; i < BATCHES * 16 * D; i += blockDim.x) {
        int b = i / (16 * D);
        int c = (i / D) & 15;
        int d = i % D;
        float v = 0.f;
        if (c < C) {
            float cnt = g_ccnt[b * C + c];
            v = g_fsum[((size_t)b * C + c) * D + d] / fmaxf(cnt, 1.0f);
        }
        s_cm[b][c][d] = v;
    }
    if (threadIdx.x < BATCHES) { s_inter[threadIdx.x] = 0.f; s_np[threadIdx.x] = 0.f; }
    __syncthreads();

    const int wave = threadIdx.x >> 5;  // 8 waves == 8 batches
    const int lane = threadIdx.x & 31;
    const int half = lane >> 4;
    const int m = lane & 15;
    {
        // Gram = cm . cm^T via WMMA f32 16x16x4; A and B fragments coincide
        // (lane l holds row (l&15), K = 4*k4 + 2*half + {0,1}).
        const int b = wave;
        v8f acc = {};
#pragma unroll
        for (int k4 = 0; k4 < D / 4; ++k4) {
            int kb = k4 * 4 + 2 * half;
            v2f a;
            a.x = s_cm[b][m][kb];
            a.y = s_cm[b][m][kb + 1];
            acc = __builtin_amdgcn_wmma_f32_16x16x4_f32(false, a, false, a, (short)0, acc, false, false);
        }
#pragma unroll
        for (int v = 0; v < 8; ++v) s_gram[b][v + 8 * half][m] = acc[v];
    }
    __syncthreads();

    // Hinge terms: dist^2(i,j) = G[i][i] + G[j][j] - 2 G[i][j]
    for (int idx = threadIdx.x; idx < BATCHES * C * C; idx += blockDim.x) {
        int b = idx / (C * C);
        int r = idx % (C * C);
        int i = r / C, j = r % C;
        if (i == j) continue;
        bool pi = g_ccnt[b * C + i] > 0.f;
        bool pj = g_ccnt[b * C + j] > 0.f;
        if (!(pi && pj)) continue;
        float dd2 = s_gram[b][i][i] + s_gram[b][j][j] - 2.f * s_gram[b][i][j];
        float dist = sqrtf(fmaxf(dd2, 0.f));
        float term = fmaxf(MARGIN - dist, 0.f);
        atomicAdd(&s_inter[b], term);
        atomicAdd(&s_np[b], 1.f);
    }
    __syncthreads();

    if (threadIdx.x == 0) {
        float ti = 0.f, te = 0.f, bc = 0.f;
        for (int b = 0; b < BATCHES; ++b) {
            float cnt = 0.f;
            for (int c = 0; c < C; ++c) cnt += g_ccnt[b * C + c];  // cnt_b = sum_c ccnt
            if (cnt > 0.f) {
                bc += 1.f;
                ti += g_d2[b] / fmaxf(cnt, 1.f);
                te += s_inter[b] / fmaxf(s_np[b], 1.f);
            }
        }
        float denom = fmaxf(bc, 1.f);
        out[0] = LOSS_W * (ti / denom + te / denom);
    }
}

extern "C" void kernel_launch(void* const* d_in, const int* in_sizes, int n_in,
                              void* d_out, int out_size, void* d_ws, size_t ws_size,
                              hipStream_t stream) {
    // inputs: 0=pred(N*C, unused), 1=target(N,i32), 2=feat(N*D,f32), 3=batch(N,i32), 4=centers(C*D,f32)
    const int* target = (const int*)d_in[1];
    const float* feat = (const float*)d_in[2];
    const int* batch = (const int*)d_in[3];
    const float* centers = (const float*)d_in[4];
    float* out = (float*)d_out;
    const int N = in_sizes[1];

    float* ws = (float*)d_ws;
    float* g_fsum = ws;                        // BATCHES*C*D
    float* g_ccnt = g_fsum + BATCHES * C * D;  // BATCHES*C
    float* g_d2 = g_ccnt + BATCHES * C;        // BATCHES
    const int wsFloats = BATCHES * C * D + BATCHES * C + BATCHES;

    dcl_zero_ws<<<(wsFloats + 255) / 256, 256, 0, stream>>>(ws, wsFloats);

    const int blocks = 512;                    // 4096 waves: enough MLP for 23.3 TB/s
    const int totalWaves = blocks * (256 / 32);
    int ppw = (N + totalWaves - 1) / totalWaves;
    ppw = (ppw + 15) & ~15;                    // supergroup multiple (keeps int4 loads aligned)
    dcl_phase1<<<blocks, 256, 0, stream>>>(target, feat, batch, centers,
                                           g_fsum, g_ccnt, g_d2, N, ppw);
    dcl_finalize<<<1, 256, 0, stream>>>(g_fsum, g_ccnt, g_d2, out);
}